// Mse_Aff_20126216749203
// MI455X (gfx1250) — compile-verified
//
#include <hip/hip_runtime.h>
#include <math.h>

#define IGNORE_LB 255
#define NCLS 19
#define PH 96
#define PW 96
#define N_AFF (PH * PW)              /* 9216 */
#define HH 768
#define WW 768
#define ROWS_PER_BLK 16
#define AFF_BLOCKS (N_AFF / ROWS_PER_BLK)   /* 576 */
#define CE_BLOCKS ((HH * WW) / 256)         /* 2304 */

typedef __attribute__((ext_vector_type(2))) float v2f;
typedef __attribute__((ext_vector_type(8))) float v8f;

// ---------------------------------------------------------------------------
// ws layout (floats):
//   [0, 9216)            lab (downsampled labels, as float)
//   [9216, 9216+576)     aff per-block partial sums of (pred-gt)^2
//   [9792, 9792+3*2304)  ce per-block partials: {n1, n2, d} interleaved
// ---------------------------------------------------------------------------

__global__ void labels_kernel(const int* __restrict__ target, float* __restrict__ lab) {
    int i = blockIdx.x * blockDim.x + threadIdx.x;
    if (i < N_AFF) {
        int r = i / PW, c = i % PW;
        // src_idx = floor(dst*in/out) = dst*8 exactly (768/96)
        lab[i] = (float)target[(r * 8) * WW + c * 8];
    }
}

// Streams aff_pred (340 MB) once; squared errors are accumulated through the
// matrix unit: V_WMMA_F32_16X16X4_F32 with an all-ones B folds 64 values/issue
// into the persistent 16x16 f32 accumulator C.
__global__ void aff_kernel(const float* __restrict__ pred,
                           const float* __restrict__ lab,
                           float* __restrict__ partials) {
    __shared__ __align__(16) float labf[N_AFF];
    __shared__ float wsum[8];

    const int tid = threadIdx.x;
    for (int i = tid; i < N_AFF; i += 256) labf[i] = lab[i];
    __syncthreads();

    const int wid  = tid >> 5;
    const int lane = tid & 31;
    const int row0 = blockIdx.x * ROWS_PER_BLK + wid * 2;  // 2 rows per wave

    v2f bones; bones.x = 1.0f; bones.y = 1.0f;
    v8f c = {};  // persistent accumulator (C matrix)

    for (int rr = 0; rr < 2; ++rr) {
        const int r = row0 + rr;
        const float lr = labf[r];
        const float* __restrict__ prow = pred + (size_t)r * N_AFF;
        for (int cb = 0; cb < N_AFF; cb += 64) {          // 144 iters, uniform
            const int col = cb + lane * 2;
            const float2 p  = *(const float2*)(prow + col);   // coalesced b64
            const float2 lb = *(const float2*)(labf + col);   // LDS b64
            float g0 = (lb.x == lr) ? 1.0f : 0.0f;
            float g1 = (lb.y == lr) ? 1.0f : 0.0f;
            float e0 = p.x - g0;
            float e1 = p.y - g1;
            v2f a; a.x = e0 * e0; a.y = e1 * e1;
            // D[m][n] = sum_k A[m][k] * 1 + C[m][n]
            c = __builtin_amdgcn_wmma_f32_16x16x4_f32(
                    false, a, false, bones, (short)0, c, false, false);
        }
    }

    // Every column of C is identical (B was all-ones).
    // lanes 0-15: s = sum of rows 0-7; lanes 16-31: rows 8-15.
    float s = c[0] + c[1] + c[2] + c[3] + c[4] + c[5] + c[6] + c[7];
    float tot = s + __shfl_xor(s, 16, 32);
    if (lane == 0) wsum[wid] = tot;
    __syncthreads();
    if (tid == 0) {
        float b = 0.0f;
        for (int i = 0; i < 8; ++i) b += wsum[i];
        partials[blockIdx.x] = b;
    }
}

// One output pixel per thread: half-pixel bilinear (edge-clamped) sample of
// both logit maps, 19-way log-softmax, weighted NLL. Logits are L2-resident.
__global__ void ce_kernel(const float* __restrict__ lg1,
                          const float* __restrict__ lg2,
                          const float* __restrict__ cw,
                          const int* __restrict__ target,
                          float* __restrict__ partials) {
    const int tid = threadIdx.x;
    const int pix = blockIdx.x * 256 + tid;   // grid covers exactly 768*768
    const int y = pix / WW, x = pix % WW;

    float sy = (y + 0.5f) * 0.125f - 0.5f;
    float sx = (x + 0.5f) * 0.125f - 0.5f;
    float fy = floorf(sy), fx = floorf(sx);
    float wy = sy - fy,    wx = sx - fx;
    int iy = (int)fy, ix = (int)fx;
    int y0 = min(PH - 1, max(0, iy));
    int y1 = min(PH - 1, max(0, iy + 1));
    int x0 = min(PW - 1, max(0, ix));
    int x1 = min(PW - 1, max(0, ix + 1));
    float w00 = (1.0f - wy) * (1.0f - wx), w01 = (1.0f - wy) * wx;
    float w10 = wy * (1.0f - wx),          w11 = wy * wx;
    int i00 = y0 * PW + x0, i01 = y0 * PW + x1;
    int i10 = y1 * PW + x0, i11 = y1 * PW + x1;

    int t = target[pix];
    float n1 = 0.0f, n2 = 0.0f, d = 0.0f;
    if (t != IGNORE_LB) {
        float w = cw[t];
        float v1[NCLS], v2[NCLS];
        float m1 = -1e30f, m2 = -1e30f;
        float pick1 = 0.0f, pick2 = 0.0f;
#pragma unroll
        for (int ch = 0; ch < NCLS; ++ch) {
            const float* p = lg1 + ch * N_AFF;
            float v = w00 * p[i00] + w01 * p[i01] + w10 * p[i10] + w11 * p[i11];
            v1[ch] = v; m1 = fmaxf(m1, v);
            const float* q = lg2 + ch * N_AFF;
            float u = w00 * q[i00] + w01 * q[i01] + w10 * q[i10] + w11 * q[i11];
            v2[ch] = u; m2 = fmaxf(m2, u);
            pick1 = (ch == t) ? v : pick1;
            pick2 = (ch == t) ? u : pick2;
        }
        float s1 = 0.0f, s2 = 0.0f;
#pragma unroll
        for (int ch = 0; ch < NCLS; ++ch) {
            s1 += __expf(v1[ch] - m1);
            s2 += __expf(v2[ch] - m2);
        }
        float lse1 = m1 + __logf(s1);
        float lse2 = m2 + __logf(s2);
        n1 = w * (lse1 - pick1);
        n2 = w * (lse2 - pick2);
        d  = w;
    }

    __shared__ float sn1[256], sn2[256], sd[256];
    sn1[tid] = n1; sn2[tid] = n2; sd[tid] = d;
    __syncthreads();
    for (int off = 128; off > 0; off >>= 1) {
        if (tid < off) {
            sn1[tid] += sn1[tid + off];
            sn2[tid] += sn2[tid + off];
            sd[tid]  += sd[tid + off];
        }
        __syncthreads();
    }
    if (tid == 0) {
        partials[3 * blockIdx.x + 0] = sn1[0];
        partials[3 * blockIdx.x + 1] = sn2[0];
        partials[3 * blockIdx.x + 2] = sd[0];
    }
}

__global__ void final_kernel(const float* __restrict__ aff_p,
                             const float* __restrict__ ce_p,
                             float* __restrict__ out) {
    const int tid = threadIdx.x;
    float a = 0.0f, n1 = 0.0f, n2 = 0.0f, d = 0.0f;
    for (int i = tid; i < AFF_BLOCKS; i += 256) a += aff_p[i];
    for (int i = tid; i < CE_BLOCKS; i += 256) {
        n1 += ce_p[3 * i + 0];
        n2 += ce_p[3 * i + 1];
        d  += ce_p[3 * i + 2];
    }
    __shared__ float sa[256], s1[256], s2[256], sd[256];
    sa[tid] = a; s1[tid] = n1; s2[tid] = n2; sd[tid] = d;
    __syncthreads();
    for (int off = 128; off > 0; off >>= 1) {
        if (tid < off) {
            sa[tid] += sa[tid + off];
            s1[tid] += s1[tid + off];
            s2[tid] += s2[tid + off];
            sd[tid] += sd[tid + off];
        }
        __syncthreads();
    }
    if (tid == 0) {
        float aff_loss = sa[0] / ((float)N_AFF * (float)N_AFF);
        float loss1 = s1[0] / sd[0];
        float loss2 = s2[0] / sd[0];
        out[0] = 0.4f * loss1 + loss2 + 1.0f * aff_loss;   // DSN_W, AFF_W
    }
}

extern "C" void kernel_launch(void* const* d_in, const int* in_sizes, int n_in,
                              void* d_out, int out_size, void* d_ws, size_t ws_size,
                              hipStream_t stream) {
    const float* aff_pred = (const float*)d_in[0];  // [1, 9216, 9216]
    const float* logits1  = (const float*)d_in[1];  // [1, 19, 96, 96]
    const float* logits2  = (const float*)d_in[2];  // [1, 19, 96, 96]
    const float* cweight  = (const float*)d_in[3];  // [19]
    const int*   target   = (const int*)d_in[4];    // [1, 768, 768]
    float* out = (float*)d_out;

    float* ws    = (float*)d_ws;
    float* lab   = ws;                  // 9216
    float* aff_p = ws + N_AFF;          // 576
    float* ce_p  = aff_p + AFF_BLOCKS;  // 3*2304

    labels_kernel<<<(N_AFF + 255) / 256, 256, 0, stream>>>(target, lab);
    aff_kernel<<<AFF_BLOCKS, 256, 0, stream>>>(aff_pred, lab, aff_p);
    ce_kernel<<<CE_BLOCKS, 256, 0, stream>>>(logits1, logits2, cweight, target, ce_p);
    final_kernel<<<1, 256, 0, stream>>>(aff_p, ce_p, out);
}